// CannyDetector_21526376087662
// MI455X (gfx1250) — compile-verified
//
#include <hip/hip_runtime.h>
#include <hip/hip_bf16.h>

typedef __attribute__((ext_vector_type(2)))  float v2f;
typedef __attribute__((ext_vector_type(8)))  float v8f;

#define IMG   512
#define TILE  32
#define CH    8
#define CIN   3

// staged window widths (output tile + halo)
#define W0 40            // h  (tile+8)
#define N0 (W0*W0)       // 1600
#define W1 38            // u / g2 / g3 (tile+6)
#define N1 (W1*W1)       // 1444
#define W2 36            // mag / s / idx (tile+4)
#define N2 (W2*W2)       // 1296
#define W3 34            // m  (tile+2)
#define N3 (W3*W3)       // 1156

// ---- LDS float offsets, with lifetime-based aliasing ----
// Region A [0..12800): shH (stages 1-2a)  -> shG2 (stage 2b+) + shM tail
#define F_H    0                    // 12800 floats
#define F_G2   0                    // 11552 floats (aliases dead shH)
#define F_M    11552                // 1156  floats (A-region slack)
// Region B [12800..24464): shU (stages 2a-2b) -> shMag + shS (stage 3+)
#define F_U    12800                // 11552 floats
#define F_MAG  12800                // 10368 floats (aliases dead shU)
#define F_S    (F_MAG + CH*N2)      // 1296 floats  (23168..24464)
// Region C [24464..36016): shG3
#define F_G3   24464                // 11552 floats
// weights
#define F_QX   36016                // 64
#define F_QY   (F_QX + 64)          // 64
#define F_MRG  (F_QY + 64)          // 8
#define F_LOW  (F_MRG + 8)          // 8
#define F_HIGH (F_LOW + 8)          // 8
#define F_END  (F_HIGH + 8)         // 36168 floats
#define SMEM_BYTES ((size_t)(F_END*4 + CH*N2))   // + u8 idx tail = 155,040 B (<160KB -> 2 WG/WGP)

__device__ __forceinline__ int reflect1(int i, int n) {
    i = (i < 0) ? -i : i;
    i = (i >= n) ? (2*n - 2 - i) : i;
    return i;
}

__device__ __forceinline__ float fast_sig(float z) {
    return 1.0f / (1.0f + __expf(-z));
}

__device__ __forceinline__ v8f wmma4(v2f a, v2f b, v8f c) {
    return __builtin_amdgcn_wmma_f32_16x16x4_f32(false, a, false, b, (short)0, c, false, false);
}

__global__ __launch_bounds__(256)
void canny_fused_kernel(const float* __restrict__ x,
                        const float* __restrict__ wexp,
                        const float* __restrict__ wpg,
                        const float* __restrict__ wpx,
                        const float* __restrict__ wpy,
                        const float* __restrict__ wmrg,
                        const float* __restrict__ lowt,
                        const float* __restrict__ hight,
                        float* __restrict__ out)
{
    extern __shared__ float smem[];
    float* shH   = smem + F_H;
    float* shG2  = smem + F_G2;
    float* shG3  = smem + F_G3;
    float* shU   = smem + F_U;
    float* shMag = smem + F_MAG;
    float* shS   = smem + F_S;
    float* shM   = smem + F_M;
    float* shQX  = smem + F_QX;
    float* shQY  = smem + F_QY;
    float* shMRG = smem + F_MRG;
    float* shLOW = smem + F_LOW;
    float* shHIGH= smem + F_HIGH;
    unsigned char* shIdx = (unsigned char*)(smem + F_END);

    const int tid = threadIdx.x;
    const int tx0 = blockIdx.x * TILE;
    const int ty0 = blockIdx.y * TILE;
    const int b   = blockIdx.z;
    const float* xb = x + (size_t)b * CIN * IMG * IMG;

    // ---- Stage 0: fold permutes through the (linear, depthwise) sobel:
    //      Qx = P_sx * P_g,  Qy = P_sy * P_g   (reads globals only -> no barrier needed)
    if (tid < 64) {
        const int o = tid >> 3, k = tid & 7;
        float qx = 0.0f, qy = 0.0f;
        #pragma unroll
        for (int j = 0; j < 8; ++j) {
            const float pg = wpg[j*8 + k];
            qx += wpx[o*8 + j] * pg;
            qy += wpy[o*8 + j] * pg;
        }
        shQX[tid] = qx;
        shQY[tid] = qy;
    }
    if (tid < 8) { shMRG[tid] = wmrg[tid]; shLOW[tid] = lowt[tid]; shHIGH[tid] = hight[tid]; }

    // ================= Stage 1: 1x1 expand (3 -> 8 ch) via f32 WMMA =================
    const int ln   = tid & 31;
    const int wave = tid >> 5;
    const int row  = ln & 15;
    const int hi   = (ln >> 4) & 1;          // lanes 16..31 carry K=+2/+3
    const int kA   = hi ? 2 : 0;             // K slot in VGPR0
    const int kB   = hi ? 3 : 1;             // K slot in VGPR1

    {
        v2f A;
        A.x = (row < CH) ? ((kA < CIN) ? wexp[row*3 + kA] : 0.0f) : 0.0f;
        A.y = (row < CH) ? ((kB < CIN) ? wexp[row*3 + kB] : 0.0f) : 0.0f;
        for (int grp = wave; grp < N0/16; grp += 8) {        // wave-uniform, EXEC all-1s
            const int p  = grp*16 + (ln & 15);               // 0..1599
            const int ry = p / W0, rx = p - ry*W0;
            const int iy = min(max(ty0 - 4 + ry, 0), IMG - 1);
            const int ix = min(max(tx0 - 4 + rx, 0), IMG - 1);
            const size_t pix = (size_t)iy * IMG + ix;
            v2f B;
            B.x = hi ? xb[2*(size_t)IMG*IMG + pix] : xb[pix];
            B.y = hi ? 0.0f                        : xb[1*(size_t)IMG*IMG + pix];
            v8f C = {};
            v8f D = wmma4(A, B, C);
            if (!hi) {
                #pragma unroll
                for (int c = 0; c < CH; ++c) shH[c*N0 + p] = D[c];
            }
        }
    }
    __syncthreads();

    // ============ Stage 2a: un-permuted gaussian blur (reflect) -> shU ============
    const float GW0 = 0.11957165f;   // center  1/S
    const float GW1 = 0.11310993f;   // edge    exp(-1/18)/S
    const float GW2 = 0.10699704f;   // corner  exp(-1/9)/S
    for (int e = tid; e < N1; e += 256) {
        const int ry = e / W1, rx = e - ry*W1;
        const int iy = ty0 - 3 + ry, ix = tx0 - 3 + rx;
        if (iy < 0 || iy >= IMG || ix < 0 || ix >= IMG) continue;   // slot never read
        const int ym = reflect1(iy-1, IMG) - (ty0-4);
        const int yc = iy - (ty0-4);
        const int yp = reflect1(iy+1, IMG) - (ty0-4);
        const int xm = reflect1(ix-1, IMG) - (tx0-4);
        const int xc = ix - (tx0-4);
        const int xp = reflect1(ix+1, IMG) - (tx0-4);
        #pragma unroll
        for (int k = 0; k < CH; ++k) {
            const float* hk = shH + k*N0;
            shU[k*N1 + e] =
                  GW2*(hk[ym*W0+xm] + hk[ym*W0+xp] + hk[yp*W0+xm] + hk[yp*W0+xp])
                + GW1*(hk[ym*W0+xc] + hk[yc*W0+xm] + hk[yc*W0+xp] + hk[yp*W0+xc])
                + GW0* hk[yc*W0+xc];
        }
    }
    __syncthreads();   // shU complete; shH now dead (shG2 may overwrite region A)

    // ======= Stage 2b: WMMA double-GEMM  G2 = Qx*U, G3 = Qy*U  (8x8 x 8xN1) =======
    {
        v2f AX0, AX1, AY0, AY1;
        const bool rv = (row < CH);
        AX0.x = rv ? shQX[row*8 + 0 + kA] : 0.0f;
        AX0.y = rv ? shQX[row*8 + 0 + kB] : 0.0f;
        AX1.x = rv ? shQX[row*8 + 4 + kA] : 0.0f;
        AX1.y = rv ? shQX[row*8 + 4 + kB] : 0.0f;
        AY0.x = rv ? shQY[row*8 + 0 + kA] : 0.0f;
        AY0.y = rv ? shQY[row*8 + 0 + kB] : 0.0f;
        AY1.x = rv ? shQY[row*8 + 4 + kA] : 0.0f;
        AY1.y = rv ? shQY[row*8 + 4 + kB] : 0.0f;

        for (int grp = wave; grp < (N1 + 15)/16; grp += 8) {   // 91 groups, wave-uniform
            const int p = min(grp*16 + (ln & 15), N1 - 1);     // dup writes are same-value
            v2f B0, B1;                                        // K rows 0..3 and 4..7
            B0.x = shU[(0 + kA)*N1 + p];
            B0.y = shU[(0 + kB)*N1 + p];
            B1.x = shU[(4 + kA)*N1 + p];
            B1.y = shU[(4 + kB)*N1 + p];
            v8f Z = {};
            v8f Dx = wmma4(AX0, B0, Z);
            Dx     = wmma4(AX1, B1, Dx);
            v8f Dy = wmma4(AY0, B0, Z);
            Dy     = wmma4(AY1, B1, Dy);
            if (!hi) {
                #pragma unroll
                for (int c = 0; c < CH; ++c) {
                    shG2[c*N1 + p] = Dx[c];
                    shG3[c*N1 + p] = Dy[c];
                }
            }
        }
    }
    __syncthreads();   // shG2/shG3 complete; shU dead (shMag/shS overwrite region B)

    // ====== Stage 3: 6-tap sobel per channel -> mag, channel-sum s, dir idx ======
    for (int e = tid; e < N2; e += 256) {
        const int ry = e / W2, rx = e - ry*W2;
        const int iy = ty0 - 2 + ry, ix = tx0 - 2 + rx;
        if (iy < 0 || iy >= IMG || ix < 0 || ix >= IMG) { shS[e] = 0.0f; continue; } // zero-pad for dirs
        const int ym = reflect1(iy-1, IMG) - (ty0-3);
        const int yc = iy - (ty0-3);
        const int yp = reflect1(iy+1, IMG) - (ty0-3);
        const int xm = reflect1(ix-1, IMG) - (tx0-3);
        const int xc = ix - (tx0-3);
        const int xp = reflect1(ix+1, IMG) - (tx0-3);
        float ssum = 0.0f;
        #pragma unroll
        for (int o = 0; o < CH; ++o) {
            const float* g2 = shG2 + o*N1;
            const float* g3 = shG3 + o*N1;
            const float gx = 0.5f*(g2[ym*W1+xp] - g2[ym*W1+xm])
                           +      (g2[yc*W1+xp] - g2[yc*W1+xm])
                           + 0.5f*(g2[yp*W1+xp] - g2[yp*W1+xm]);
            const float gy = 0.5f*(g3[yp*W1+xm] - g3[ym*W1+xm])
                           +      (g3[yp*W1+xc] - g3[ym*W1+xc])
                           + 0.5f*(g3[yp*W1+xp] - g3[ym*W1+xp]);
            const float mag = sqrtf(gx*gx + gy*gy + 1e-10f);
            shMag[o*N2 + e] = mag;
            ssum += mag;
            const float ang = atan2f(gy, gx) * 1.27323954474f;   // deg/45
            shIdx[o*N2 + e] = (unsigned char)(((int)rintf(ang)) & 7);
        }
        shS[e] = ssum;
    }
    __syncthreads();

    // ===== Stage 4: NMS (dirs = s(p) - s_zp(p+off)), double threshold, merge -> m =====
    const int offy[8] = { 0, 1, 1, 1, 0,-1,-1,-1 };
    const int offx[8] = { 1, 1, 0,-1,-1,-1, 0, 1 };
    for (int e = tid; e < N3; e += 256) {
        const int ry = e / W3, rx = e - ry*W3;
        const int iy = ty0 - 1 + ry, ix = tx0 - 1 + rx;
        if (iy < 0 || iy >= IMG || ix < 0 || ix >= IMG) continue;    // slot never read
        const int sy = iy - (ty0-2), sx = ix - (tx0-2);
        const float sc = shS[sy*W2 + sx];
        float m = 0.0f;
        #pragma unroll
        for (int o = 0; o < CH; ++o) {
            const int id  = shIdx[o*N2 + sy*W2 + sx];
            const int id2 = (id + 4) & 7;
            int qy = iy + offy[id],  qx = ix + offx[id];
            const float sp = (qy>=0 && qy<IMG && qx>=0 && qx<IMG)
                           ? shS[(qy-(ty0-2))*W2 + (qx-(tx0-2))] : 0.0f;
            qy = iy + offy[id2];  qx = ix + offx[id2];
            const float sn = (qy>=0 && qy<IMG && qx>=0 && qx<IMG)
                           ? shS[(qy-(ty0-2))*W2 + (qx-(tx0-2))] : 0.0f;
            const float pos = sc - sp, neg = sc - sn;
            const float mag = shMag[o*N2 + sy*W2 + sx];
            const float nms = (pos > 0.0f && neg > 0.0f) ? mag : 0.0f;
            const float ht = shHIGH[o], lt = shLOW[o];
            const float strong = fast_sig(1000.0f * (nms - ht));
            const float weak   = 0.5f * (fast_sig(1000.0f * (ht - nms))
                                       - fast_sig(1000.0f * (lt - nms)));
            m += shMRG[o] * (weak + strong);
        }
        shM[e] = m;
    }
    __syncthreads();

    // ===== Stage 5: one-step hysteresis (reflect) + write output =====
    for (int e = tid; e < TILE*TILE; e += 256) {
        const int ry = e >> 5, rx = e & 31;
        const int iy = ty0 + ry, ix = tx0 + rx;
        const float m0 = shM[(ry+1)*W3 + (rx+1)];
        const float strong0 = (m0 == 1.0f) ? m0 : 0.0f;
        float hyst = 0.0f;
        #pragma unroll
        for (int dy = -1; dy <= 1; ++dy) {
            #pragma unroll
            for (int dx = -1; dx <= 1; ++dx) {
                const int qy = reflect1(iy + dy, IMG);
                const int qx = reflect1(ix + dx, IMG);
                const float mq = shM[(qy-(ty0-1))*W3 + (qx-(tx0-1))];
                hyst += 1.25f * ((mq == 1.0f) ? mq : 0.0f);
            }
        }
        const float res = (((hyst > 1.0f) && (m0 == 0.5f)) ? m0 : 0.0f) + strong0;
        out[((size_t)b*IMG + iy)*IMG + ix] = res;
    }
}

extern "C" void kernel_launch(void* const* d_in, const int* in_sizes, int n_in,
                              void* d_out, int out_size, void* d_ws, size_t ws_size,
                              hipStream_t stream) {
    (void)in_sizes; (void)n_in; (void)d_ws; (void)ws_size; (void)out_size;
    const float* x     = (const float*)d_in[0];
    const float* wexp  = (const float*)d_in[1];
    const float* wpg   = (const float*)d_in[2];
    const float* wpx   = (const float*)d_in[3];
    const float* wpy   = (const float*)d_in[4];
    const float* wmrg  = (const float*)d_in[5];
    const float* lowt  = (const float*)d_in[6];
    const float* hight = (const float*)d_in[7];
    float* out = (float*)d_out;

    dim3 grid(IMG/TILE, IMG/TILE, 8);   // 16 x 16 tiles x B=8
    dim3 block(256);                    // 8 wave32
    canny_fused_kernel<<<grid, block, SMEM_BYTES, stream>>>(
        x, wexp, wpg, wpx, wpy, wmrg, lowt, hight, out);
}